// ChamferLossSplitPID_81423989997794
// MI455X (gfx1250) — compile-verified
//
#include <hip/hip_runtime.h>

// ChamferLossSplitPID on MI455X (gfx1250):
//   dist(i,j)^2 = |t_i|^2 + |r_j|^2 - 2 t_i.r_j
// Cross terms via V_WMMA_F32_16X16X4_F32 (K = D = 4, full fp32).
// One workgroup per batch; working set staged in LDS.
// - Column packet jpack[half][j] = {r[k0], r[k0+1], nr^2, pid}: ONE
//   ds_load_b128 per tile, software-pipelined one tile ahead so no
//   s_wait_dscnt sits in front of the WMMA.
// - Phase-2 only reads rowmin[i] at p=in_pid[i] / colmin[j] at p=out_pid[j];
//   both reductions share the gate (out_pid[j]==in_pid[i]) -> one biased
//   value feeds a scalar row-min (registers) and a scalar col-min (ds_min_u32).
// - Mins tracked in squared-distance space; sqrt once per particle at the end.

typedef __attribute__((ext_vector_type(2))) float v2f;
typedef __attribute__((ext_vector_type(8))) float v8f;

#define NB 64
#define NP 512
#define NT 256          // 8 waves of 32
#define INF_U 0x7F800000u

__global__ void chamfer_init_out(float* out) {
    if (threadIdx.x < 2) out[threadIdx.x] = 0.0f;
}

__global__ __launch_bounds__(NT)
void chamfer_kernel(const float* __restrict__ target,
                    const float* __restrict__ reco,
                    const int* __restrict__ in_pid,
                    const int* __restrict__ out_pid,
                    float* __restrict__ out) {
    __shared__ float  t_lds[NP * 4];      // target[b] row-major [i][k]  (A fragments)
    __shared__ float4 jpack[2 * NP];      // [khalf][j] = {r[k0], r[k0+1], nr^2, as_float(out_pid)}
    __shared__ float2 tmeta[NP];          // {|t_i|^2, as_float(in_pid==0 ? -1 : in_pid)}
    __shared__ float  nt_lds[NP];         // ||t_i||   (Phase 2 sums)
    __shared__ float  nr_lds[NP];         // ||r_j||
    __shared__ int    ip_lds[NP];         // in_pid    (Phase 2 indexing)
    __shared__ int    op_lds[NP];         // out_pid
    __shared__ unsigned rowmin[NP];       // min_j d^2 over {j: out_pid[j]==in_pid[i]}
    __shared__ unsigned colmin[NP];       // min_i d^2 over {i: in_pid[i]==out_pid[j]}
    __shared__ float s_sum_nt[5], s_sum_nr[5], s_sum_xy[4], s_sum_yx[4];
    __shared__ int   s_cnt_in[5], s_cnt_out[5];

    const int b   = blockIdx.x;
    const int tid = threadIdx.x;
    const float INF = __uint_as_float(INF_U);

    const float* tb  = target + (size_t)b * NP * 4;
    const float* rb  = reco   + (size_t)b * NP * 4;
    const int*   ipb = in_pid  + b * NP;
    const int*   opb = out_pid + b * NP;

    // ---- Phase 0: stage to LDS, compute norms/metadata, init reductions ----
    for (int i = tid; i < NP; i += NT) {
        float4 t = ((const float4*)tb)[i];
        float4 r = ((const float4*)rb)[i];
        ((float4*)t_lds)[i] = t;
        const float nt2 = t.x * t.x + t.y * t.y + t.z * t.z + t.w * t.w;
        const float nr2 = r.x * r.x + r.y * r.y + r.z * r.z + r.w * r.w;
        nt_lds[i] = __builtin_amdgcn_sqrtf(nt2);
        nr_lds[i] = __builtin_amdgcn_sqrtf(nr2);
        const int pi = ipb[i];
        const int po = opb[i];
        ip_lds[i] = pi;
        op_lds[i] = po;
        const float pof = __int_as_float(po);
        jpack[i]      = make_float4(r.x, r.y, nr2, pof);   // khalf 0: K={0,1}
        jpack[NP + i] = make_float4(r.z, r.w, nr2, pof);   // khalf 1: K={2,3}
        tmeta[i] = make_float2(nt2, __int_as_float(pi == 0 ? -1 : pi));
        rowmin[i] = INF_U;
        colmin[i] = INF_U;
    }
    if (tid < 5) { s_cnt_in[tid] = 0; s_cnt_out[tid] = 0; s_sum_nt[tid] = 0.f; s_sum_nr[tid] = 0.f; }
    if (tid < 4) { s_sum_xy[tid] = 0.f; s_sum_yx[tid] = 0.f; }
    __syncthreads();

    // ---- Phase 1: tiled squared-distance matrix via V_WMMA_F32_16X16X4_F32 ----
    const int lane  = tid & 31;
    const int wave  = tid >> 5;
    const int n_loc = lane & 15;          // A: M index / B: N index / C,D: N index
    const int khalf = lane >> 4;          // 0 or 1
    const int k0    = khalf * 2;          // A/B K split: K={0,1} or {2,3}
    const int mbase = khalf * 8;          // C/D M split: rows v or v+8

    const float4* jp = &jpack[khalf * NP];  // this lane's half-specific packets

#pragma clang loop unroll(disable)
    for (int I = wave; I < NP / 16; I += NT / 32) {
        // A fragment: target rows of tile I (16x4 fp32, 2 VGPRs/lane)
        const int irow = I * 16 + n_loc;
        const float2 af = *(const float2*)&t_lds[irow * 4 + k0];
        v2f a; a.x = af.x; a.y = af.y;

        // Row metadata for this lane's 8 C/D rows
        float ntm[8]; int pimx[8];
#pragma unroll
        for (int v = 0; v < 8; ++v) {
            const float2 tm = tmeta[I * 16 + mbase + v];
            ntm[v]  = tm.x;
            pimx[v] = __float_as_int(tm.y);   // in {-1,1,2,3,4}
        }

        // Scalar per-row mins (squared distances) across the J sweep
        float rmin[8];
#pragma unroll
        for (int v = 0; v < 8; ++v) rmin[v] = INF;

        // Software pipeline: packet for tile J is loaded one iteration ahead.
        float4 cur = jp[n_loc];               // J = 0
#pragma clang loop unroll_count(4)
        for (int J = 0; J < NP / 16; ++J) {
            const int jcol = J * 16 + n_loc;
            v2f bv; bv.x = cur.x; bv.y = cur.y;
            const float nr2 = cur.z;
            const int   pj  = __float_as_int(cur.w);  // {0..4}; 0 never matches pimx

            // Prefetch next tile's packet (branchless wrap keeps it in-bounds);
            // its dscnt wait lands behind the WMMA + epilogue below.
            const float4 nxt = jp[((J + 1) & (NP / 16 - 1)) * 16 + n_loc];

            v8f c = {0.f, 0.f, 0.f, 0.f, 0.f, 0.f, 0.f, 0.f};
            c = __builtin_amdgcn_wmma_f32_16x16x4_f32(
                    false, a, false, bv, (short)0, c, false, false);

            float best = INF;             // column min for this lane's column
#pragma unroll
            for (int v = 0; v < 8; ++v) {
                const float d2 = fmaxf(ntm[v] + nr2 - 2.0f * c[v], 0.0f);
                const float biased = (pj == pimx[v]) ? d2 : INF;  // shared gate
                rmin[v] = fminf(rmin[v], biased);
                best    = fminf(best, biased);
            }
            // Unconditional: atomicMin with INF is a no-op; avoids EXEC juggling.
            // (u32 order == fp32 order for non-negative values.)
            atomicMin(&colmin[jcol], __float_as_uint(best));

            cur = nxt;
        }

        // Collapse column dimension: butterfly min within each 16-lane half
        // (lanes of a half all hold the same 8 rows, different columns).
#pragma unroll
        for (int v = 0; v < 8; ++v) {
            float x = rmin[v];
            x = fminf(x, __shfl_xor(x, 1, 32));
            x = fminf(x, __shfl_xor(x, 2, 32));
            x = fminf(x, __shfl_xor(x, 4, 32));
            x = fminf(x, __shfl_xor(x, 8, 32));
            if (n_loc == 0)   // one writer per row; row tiles are wave-private
                rowmin[I * 16 + mbase + v] = __float_as_uint(x);
        }
    }
    __syncthreads();

    // ---- Phase 2: counts and masked sums (sqrt applied here, once per entry) ----
    for (int i = tid; i < NP; i += NT) {
        const int pi = ip_lds[i];
        const int po = op_lds[i];
        atomicAdd(&s_cnt_in[pi], 1);
        atomicAdd(&s_cnt_out[po], 1);
        atomicAdd(&s_sum_nt[pi], nt_lds[i]);
        atomicAdd(&s_sum_nr[po], nr_lds[i]);
        if (pi != 0) {
            const unsigned bits = rowmin[i];
            if (bits != INF_U)
                atomicAdd(&s_sum_xy[pi - 1], __builtin_amdgcn_sqrtf(__uint_as_float(bits)));
        }
        if (po != 0) {
            const unsigned bits = colmin[i];
            if (bits != INF_U)
                atomicAdd(&s_sum_yx[po - 1], __builtin_amdgcn_sqrtf(__uint_as_float(bits)));
        }
    }
    __syncthreads();

    // ---- Phase 3: per-batch finalize, mean over batch via global atomics ----
    if (tid == 0) {
        const float invB = 1.0f / (float)NB;
        const int c0 = s_cnt_out[0];
        const float lz = s_sum_nr[0] / (float)(c0 > 1 ? c0 : 1);
        atomicAdd(&out[1], lz * invB);     // loss_zero

        float lnz = 0.0f;
        for (int p = 1; p <= 4; ++p) {
            const int cx = s_cnt_in[p];
            const int cy = s_cnt_out[p];
            const float both = 0.5f * (s_sum_xy[p - 1] / (float)(cy > 1 ? cy : 1) +
                                       s_sum_yx[p - 1] / (float)(cx > 1 ? cx : 1));
            const float only_x = s_sum_nt[p] / (float)(cx > 1 ? cx : 1);
            const float only_y = s_sum_nr[p] / (float)(cy > 1 ? cy : 1);
            lnz += (cy == 0) ? only_x : ((cx == 0) ? only_y : both);
        }
        atomicAdd(&out[0], lnz * invB);    // loss_nonzero
    }
}

extern "C" void kernel_launch(void* const* d_in, const int* in_sizes, int n_in,
                              void* d_out, int out_size, void* d_ws, size_t ws_size,
                              hipStream_t stream) {
    const float* target  = (const float*)d_in[0];
    const float* reco    = (const float*)d_in[1];
    const int*   in_pid  = (const int*)d_in[2];
    const int*   out_pid = (const int*)d_in[3];
    float* out = (float*)d_out;

    hipLaunchKernelGGL(chamfer_init_out, dim3(1), dim3(32), 0, stream, out);
    hipLaunchKernelGGL(chamfer_kernel, dim3(NB), dim3(NT), 0, stream,
                       target, reco, in_pid, out_pid, out);
}